// BuildGraphPyramid_15504831939272
// MI455X (gfx1250) — compile-verified
//
#include <hip/hip_runtime.h>
#include <hip/hip_bf16.h>
#include <float.h>

// ---------------------------------------------------------------------------
// BuildGraphPyramid for MI455X (gfx1250, wave32).
//   points (B,8192,3) f32 -> FPS pyramid (8192/2048/512) + 7 k-NN (k=16) maps.
// k-NN distance tiles computed with V_WMMA_F32_16X16X4_F32 (K=3 padded to 4).
// ---------------------------------------------------------------------------

typedef __attribute__((ext_vector_type(2))) float v2f;
typedef __attribute__((ext_vector_type(8))) float v8f;

#define KNN_K 16
#define WAVES_PER_BLOCK 8   // 256 threads / 32 lanes
#define TILE_LD 18          // LDS row stride (words): 8B-aligned rows, no write conflicts

// Sorted top-16 insertion, fully register-resident (constant indices after unroll).
#define TOPK_INSERT(dval, ival)                                       \
    do {                                                              \
        float _d = (dval); int _i = (ival);                           \
        if (_d < dk[KNN_K - 1]) {                                     \
            dk[KNN_K - 1] = _d; ik[KNN_K - 1] = _i;                   \
            _Pragma("unroll")                                         \
            for (int _t = KNN_K - 1; _t > 0; --_t) {                  \
                bool  _sw = dk[_t] < dk[_t - 1];                      \
                float _fa = dk[_t - 1], _fb = dk[_t];                 \
                int   _ia = ik[_t - 1], _ib = ik[_t];                 \
                dk[_t - 1] = _sw ? _fb : _fa; dk[_t] = _sw ? _fa : _fb; \
                ik[_t - 1] = _sw ? _ib : _ia; ik[_t] = _sw ? _ia : _ib; \
            }                                                         \
        }                                                             \
    } while (0)

// ---------------------------------------------------------------------------
// Farthest point sampling: one 1024-thread workgroup per batch.
// min-dist array lives in LDS (N <= 8192 -> 32KB of the 320KB WGP pool).
// Matches reference: start at index 0, repeatedly pick argmax of min-dist.
// ---------------------------------------------------------------------------
__global__ __launch_bounds__(1024)
void fps_kernel(const float* __restrict__ pts, float* __restrict__ out,
                int N, int M) {
    __shared__ float mind[8192];
    __shared__ float wd[32];
    __shared__ int   wi[32];
    __shared__ int   s_cur;

    const int b    = blockIdx.x;
    const int tid  = threadIdx.x;
    const int lane = tid & 31;
    const int warp = tid >> 5;

    const float* p = pts + (size_t)b * N * 3;
    float*       o = out + (size_t)b * M * 3;

    for (int i = tid; i < N; i += 1024) mind[i] = FLT_MAX;

    float cx = p[0], cy = p[1], cz = p[2];
    if (tid == 0) { o[0] = cx; o[1] = cy; o[2] = cz; }
    __syncthreads();

    for (int j = 1; j < M; ++j) {
        float bd = -1.0f; int bi = 0;
        for (int i = tid; i < N; i += 1024) {
            float dx = p[3 * i + 0] - cx;
            float dy = p[3 * i + 1] - cy;
            float dz = p[3 * i + 2] - cz;
            float d  = dx * dx + dy * dy + dz * dz;
            float m  = fminf(mind[i], d);
            mind[i] = m;
            if (m > bd) { bd = m; bi = i; }
        }
        #pragma unroll
        for (int off = 16; off > 0; off >>= 1) {
            float od = __shfl_down(bd, off, 32);
            int   oi = __shfl_down(bi, off, 32);
            if (od > bd) { bd = od; bi = oi; }
        }
        if (lane == 0) { wd[warp] = bd; wi[warp] = bi; }
        __syncthreads();
        if (tid < 32) {
            bd = wd[tid]; bi = wi[tid];
            #pragma unroll
            for (int off = 16; off > 0; off >>= 1) {
                float od = __shfl_down(bd, off, 32);
                int   oi = __shfl_down(bi, off, 32);
                if (od > bd) { bd = od; bi = oi; }
            }
            if (tid == 0) s_cur = bi;
        }
        __syncthreads();
        int cur = s_cur;
        cx = p[3 * cur + 0]; cy = p[3 * cur + 1]; cz = p[3 * cur + 2];
        if (tid == 0) { o[3 * j + 0] = cx; o[3 * j + 1] = cy; o[3 * j + 2] = cz; }
        __syncthreads();
    }
}

// ---------------------------------------------------------------------------
// k-NN (k=16): one wave per 16-query tile; 8 waves per block; chunk loop
// unrolled x2 -> two back-to-back V_WMMA_F32_16X16X4_F32 per iteration.
// Selection uses ALL 32 lanes: lane r and lane r+16 each own 8 columns of
// query row (r&15) with private top-16 lists, merged once after the loop.
// Effective distance = |s|^2 - 2 q.s  (|q|^2 is row-constant: ordering-safe).
// Q multiple of 128, N multiple of 32 (holds for 8192 / 2048 / 512).
// ---------------------------------------------------------------------------
__global__ __launch_bounds__(256)
void knn16_wmma_kernel(const float* __restrict__ qpts,
                       const float* __restrict__ spts,
                       int Q, int N, int* __restrict__ out_idx) {
    __shared__ float tile[WAVES_PER_BLOCK][2][16 * TILE_LD];

    const int lane  = threadIdx.x & 31;
    const int wave  = threadIdx.x >> 5;
    const int batch = blockIdx.y;
    const int qtile = blockIdx.x * WAVES_PER_BLOCK + wave;

    const float* qb = qpts + (size_t)batch * Q * 3 + (size_t)qtile * 16 * 3;
    const float* sb = spts + (size_t)batch * N * 3;

    const int col   = lane & 15;
    const int hh    = lane >> 4;
    const int row   = col;       // query row owned during selection
    const int jbase = hh * 8;    // candidate sub-range within each 16-col tile

    // A operand: 16x4 (MxK) f32.  Lanes 0-15: (K0,K1)=(x,y) of row M=lane.
    //                             Lanes 16-31: (K2,K3)=(z,0) of row M=lane-16.
    v2f a;
    if (hh == 0) { a.x = qb[col * 3 + 0]; a.y = qb[col * 3 + 1]; }
    else         { a.x = qb[col * 3 + 2]; a.y = 0.0f; }

    float dk[KNN_K];
    int   ik[KNN_K];
    #pragma unroll
    for (int t = 0; t < KNN_K; ++t) { dk[t] = FLT_MAX; ik[t] = 0; }

    const int chunks = N >> 4;   // even for all call sites
    for (int c = 0; c < chunks; c += 2) {
        // --- two distance tiles via WMMA -------------------------------
        #pragma unroll
        for (int u = 0; u < 2; ++u) {
            const float* sc = sb + (size_t)(c + u) * 48;
            if (c + 2 < chunks)
                __builtin_prefetch((const void*)(sc + 96), 0, 1);

            // Each lane loads all 3 coords of its support column:
            // |s|^2 computed locally (no cross-lane traffic).
            const float* cp = sc + col * 3;
            float sx = cp[0], sy = cp[1], sz = cp[2];
            float s2 = sx * sx + sy * sy + sz * sz;

            // B operand: 4x16 (KxN) f32, mirrored layout of A.
            v2f bvec;
            if (hh == 0) { bvec.x = sx; bvec.y = sy; }
            else         { bvec.x = sz; bvec.y = 0.0f; }

            v8f acc = {};
            acc = __builtin_amdgcn_wmma_f32_16x16x4_f32(
                /*neg_a=*/false, a, /*neg_b=*/false, bvec,
                /*c_mod=*/(short)0, acc, /*reuse_a=*/false, /*reuse_b=*/false);

            // C layout: lane holds N=lane&15, rows M = v + 8*(lane>>4).
            #pragma unroll
            for (int v = 0; v < 8; ++v)
                tile[wave][u][(v + 8 * hh) * TILE_LD + col] = s2 - 2.0f * acc[v];
        }
        asm volatile("s_wait_dscnt 0" ::: "memory"); // intra-wave LDS RAW

        // --- selection: all 32 lanes active ----------------------------
        #pragma unroll
        for (int u = 0; u < 2; ++u) {
            #pragma unroll
            for (int jj = 0; jj < 8; ++jj) {
                int   j  = jbase + jj;
                float d  = tile[wave][u][row * TILE_LD + j];
                int   id = ((c + u) << 4) + j;
                TOPK_INSERT(d, id);
            }
        }
    }

    // Merge the two half-lists for each row: snapshot partner (shfl_xor 16),
    // then insert. Lanes 0-15 hold the final result for row = lane.
    float od[KNN_K]; int oi[KNN_K];
    #pragma unroll
    for (int t = 0; t < KNN_K; ++t) {
        od[t] = __shfl_xor(dk[t], 16, 32);
        oi[t] = __shfl_xor(ik[t], 16, 32);
    }
    #pragma unroll
    for (int t = 0; t < KNN_K; ++t) TOPK_INSERT(od[t], oi[t]);

    if (lane < 16) {
        int* orow = out_idx + ((size_t)batch * Q + (size_t)qtile * 16 + lane) * KNN_K;
        #pragma unroll
        for (int t = 0; t < KNN_K; ++t) orow[t] = ik[t];
    }
}

// ---------------------------------------------------------------------------
// Host-side launch (graph-capturable: async only, all on `stream`).
// Output layout (flat, 4-byte elements, in reference return order):
//   P0 (B,8192,3) | P1 (B,2048,3) | P2 (B,512,3) |
//   N0 (B,8192,16)| N1 (B,2048,16)| N2 (B,512,16)|
//   S0 (B,2048,16)| S1 (B,512,16) | U0 (B,8192,16)| U1 (B,2048,16)
// ---------------------------------------------------------------------------
extern "C" void kernel_launch(void* const* d_in, const int* in_sizes, int n_in,
                              void* d_out, int out_size, void* d_ws, size_t ws_size,
                              hipStream_t stream) {
    const float* pts = (const float*)d_in[0];
    const int N0n = 8192, N1n = 2048, N2n = 512;
    const int B = in_sizes[0] / (N0n * 3);

    float* P0 = (float*)d_out;
    float* P1 = P0 + (size_t)B * N0n * 3;
    float* P2 = P1 + (size_t)B * N1n * 3;
    int*   N0 = (int*)(P2 + (size_t)B * N2n * 3);
    int*   N1 = N0 + (size_t)B * N0n * KNN_K;
    int*   N2 = N1 + (size_t)B * N1n * KNN_K;
    int*   S0 = N2 + (size_t)B * N2n * KNN_K;
    int*   S1 = S0 + (size_t)B * N1n * KNN_K;
    int*   U0 = S1 + (size_t)B * N2n * KNN_K;
    int*   U1 = U0 + (size_t)B * N0n * KNN_K;

    // points_list[0] = input points
    hipMemcpyAsync(P0, pts, (size_t)B * N0n * 3 * sizeof(float),
                   hipMemcpyDeviceToDevice, stream);

    // FPS pyramid (sequential dependency via stream order)
    fps_kernel<<<B, 1024, 0, stream>>>(pts, P1, N0n, N1n);
    fps_kernel<<<B, 1024, 0, stream>>>(P1, P2, N1n, N2n);

    // self-KNN per level
    knn16_wmma_kernel<<<dim3(N0n / 128, B), 256, 0, stream>>>(pts, pts, N0n, N0n, N0);
    knn16_wmma_kernel<<<dim3(N1n / 128, B), 256, 0, stream>>>(P1,  P1,  N1n, N1n, N1);
    knn16_wmma_kernel<<<dim3(N2n / 128, B), 256, 0, stream>>>(P2,  P2,  N2n, N2n, N2);
    // subsampling: next-level queries into current-level support
    knn16_wmma_kernel<<<dim3(N1n / 128, B), 256, 0, stream>>>(P1,  pts, N1n, N0n, S0);
    knn16_wmma_kernel<<<dim3(N2n / 128, B), 256, 0, stream>>>(P2,  P1,  N2n, N1n, S1);
    // upsampling: current-level queries into next-level support
    knn16_wmma_kernel<<<dim3(N0n / 128, B), 256, 0, stream>>>(pts, P1,  N0n, N1n, U0);
    knn16_wmma_kernel<<<dim3(N1n / 128, B), 256, 0, stream>>>(P1,  P2,  N1n, N2n, U1);
}